// ResidueTF_9028021256843
// MI455X (gfx1250) — compile-verified
//
#include <hip/hip_runtime.h>
#include <hip/hip_bf16.h>
#include <math.h>

#define NN     16000
#define NE     256000
#define HID    256
#define DEG    16
#define VRAW   133
#define LOG2F_ 0.69314718055994530942f

typedef __attribute__((ext_vector_type(2))) float        v2f;
typedef __attribute__((ext_vector_type(8))) float        v8f;
typedef __attribute__((ext_vector_type(4))) unsigned int u32x4;
typedef __attribute__((ext_vector_type(8))) int          i32x8;
typedef __attribute__((ext_vector_type(4))) int          i32x4;

// TDM availability / arity detection (ROCm 7.2 = 5 args; therock-10.0 = 6 args)
#if defined(__has_builtin)
#if __has_builtin(__builtin_amdgcn_tensor_load_to_lds)
#define HAS_TDM 1
#endif
#endif
#ifndef HAS_TDM
#define HAS_TDM 0
#endif
#if __has_include(<hip/amd_detail/amd_gfx1250_TDM.h>)
#define TDM_ARGS6 1
#else
#define TDM_ARGS6 0
#endif

// AA table (21 x 6): [hydropathy, volume/100, charge, polarity, acceptor, donor]
// alphabet order: # A R N D C Q E G H I L K M F P S T W Y V
__constant__ float AA6[21 * 6] = {
    0.0f,  0.000f,  0.0f, 0.f, 0.f, 0.f,   // #
    1.8f,  0.886f,  0.0f, 0.f, 0.f, 0.f,   // A
   -4.5f,  1.734f,  1.0f, 1.f, 0.f, 1.f,   // R
   -3.5f,  1.141f,  0.0f, 1.f, 1.f, 1.f,   // N
   -3.5f,  1.111f, -1.0f, 1.f, 1.f, 0.f,   // D
    2.5f,  1.085f,  0.0f, 0.f, 0.f, 0.f,   // C
   -3.5f,  1.438f,  0.0f, 1.f, 1.f, 1.f,   // Q
   -3.5f,  1.384f, -1.0f, 1.f, 1.f, 0.f,   // E
   -0.4f,  0.601f,  0.0f, 0.f, 0.f, 0.f,   // G
   -3.2f,  1.532f,  0.1f, 1.f, 1.f, 1.f,   // H
    4.5f,  1.667f,  0.0f, 0.f, 0.f, 0.f,   // I
    3.8f,  1.667f,  0.0f, 0.f, 0.f, 0.f,   // L
   -3.9f,  1.686f,  1.0f, 1.f, 0.f, 1.f,   // K
    1.9f,  1.629f,  0.0f, 0.f, 0.f, 0.f,   // M
    2.8f,  1.899f,  0.0f, 0.f, 0.f, 0.f,   // F
   -1.6f,  1.127f,  0.0f, 0.f, 0.f, 0.f,   // P
   -0.8f,  0.890f,  0.0f, 1.f, 1.f, 1.f,   // S
   -0.7f,  1.161f,  0.0f, 1.f, 1.f, 1.f,   // T
   -0.9f,  2.278f,  0.0f, 0.f, 0.f, 1.f,   // W
   -1.3f,  1.936f,  0.0f, 1.f, 1.f, 1.f,   // Y
    4.2f,  1.400f,  0.0f, 0.f, 0.f, 0.f,   // V
};

// ---------------------------------------------------------------------------
// Node feature embedding: Vc[i, 0:133] = V[i], Vc[i, 133:256] = aa_embed(R[i])
// ---------------------------------------------------------------------------
__global__ __launch_bounds__(256) void embed_kernel(
    const float* __restrict__ V, const int* __restrict__ R,
    float* __restrict__ Vc)
{
    const int i = blockIdx.x;
    const int t = threadIdx.x;
    float out;
    if (t < VRAW) {
        out = V[(size_t)i * VRAW + t];
    } else {
        const int f = t - VRAW;
        const int r = R[i];
        const float h0 = AA6[r * 6 + 0];
        const float h1 = AA6[r * 6 + 1];
        const float h2 = AA6[r * 6 + 2];
        if (f < 90) {
            const float mu = -4.5f + (float)f * (9.0f / 89.0f);
            const float d = (h0 - mu) * 10.0f;           // stride 0.1
            out = expf(-d * d);
        } else if (f < 112) {
            const float mu = (float)(f - 90) * (2.2f / 21.0f);
            const float d = (h1 - mu) * 10.0f;           // stride 0.1
            out = expf(-d * d);
        } else if (f < 120) {
            const float mu = -1.0f + (float)(f - 112) * (2.0f / 7.0f);
            const float d = (h2 - mu) * 4.0f;            // stride 0.25
            out = expf(-d * d);
        } else {
            const float z = AA6[r * 6 + 3 + (f - 120)] * 6.0f - 3.0f;
            out = 1.0f / (1.0f + expf(-z));
        }
    }
    Vc[(size_t)i * HID + t] = out;
}

// ---------------------------------------------------------------------------
// LayerNorm over HID=256 per node (population variance, eps = 1e-5)
// ---------------------------------------------------------------------------
__global__ __launch_bounds__(256) void ln_kernel(
    const float* __restrict__ x, const float* __restrict__ w,
    const float* __restrict__ b, float* __restrict__ y)
{
    __shared__ float red[256];
    const int i = blockIdx.x;
    const int t = threadIdx.x;
    const float v = x[(size_t)i * HID + t];

    red[t] = v;
    __syncthreads();
    #pragma unroll
    for (int s = 128; s > 0; s >>= 1) {
        if (t < s) red[t] += red[t + s];
        __syncthreads();
    }
    const float mean = red[0] * (1.0f / 256.0f);
    __syncthreads();

    const float dv = v - mean;
    red[t] = dv * dv;
    __syncthreads();
    #pragma unroll
    for (int s = 128; s > 0; s >>= 1) {
        if (t < s) red[t] += red[t + s];
        __syncthreads();
    }
    const float var = red[0] * (1.0f / 256.0f);
    y[(size_t)i * HID + t] = dv * rsqrtf(var + 1e-5f) * w[t] + b[t];
}

// ---------------------------------------------------------------------------
// TDM 2-D tile load: global (row-major, f32) -> LDS with per-row padding.
// D# per CDNA5 ISA ch.7.8: group0 = {count, lds_addr, global_addr, type=2},
// group1 = {data_size, pad ctrl, tensor dims, tile dims, stride}.
// Tiles are always fully in-bounds; OOB dims only need to cover the tile.
// ---------------------------------------------------------------------------
#if HAS_TDM
__device__ __forceinline__ void tdm_load_2d(
    unsigned lds_off, const void* gptr,
    unsigned tile_d0, unsigned tile_d1,
    unsigned long long tens_d0, unsigned long long tens_d1,
    unsigned long long stride0,
    unsigned pad_interval, unsigned pad_amount)
{
    const unsigned long long ga = (unsigned long long)(uintptr_t)gptr;
    u32x4 g0;
    g0[0] = 1u;                                             // count=1, user mode
    g0[1] = lds_off;                                        // lds_addr [63:32]
    g0[2] = (unsigned)(ga & 0xFFFFFFFFull);                 // global_addr lo
    g0[3] = (unsigned)((ga >> 32) & 0x01FFFFFFull) | (2u << 30);  // hi | type=2
    i32x8 g1;
    g1[0] = (int)((2u << 16)                                // data_size = 4B
                | (1u << 20)                                // pad_enable
                | (pad_interval << 22) | (pad_amount << 25));
    g1[1] = (int)((unsigned)(tens_d0 & 0xFFFFull) << 16);   // tensor_dim0 lo16
    g1[2] = (int)(((tens_d0 >> 16) & 0xFFFFull)             // tensor_dim0 hi16
                | ((tens_d1 & 0xFFFFull) << 16));           // tensor_dim1 lo16
    g1[3] = (int)(((tens_d1 >> 16) & 0xFFFFull)             // tensor_dim1 hi16
                | (tile_d0 << 16));                         // tile_dim0
    g1[4] = (int)(tile_d1 & 0xFFFFu);                       // tile_dim1 (dim2=0)
    g1[5] = (int)(stride0 & 0xFFFFFFFFull);                 // dim0_stride lo32
    g1[6] = (int)((stride0 >> 32) & 0xFFFFull);             // dim0_stride hi16
    g1[7] = 0;
    const i32x4 z4 = {0, 0, 0, 0};
#if TDM_ARGS6
    const i32x8 z8 = {0, 0, 0, 0, 0, 0, 0, 0};
    __builtin_amdgcn_tensor_load_to_lds(g0, g1, z4, z4, z8, 0);
#else
    __builtin_amdgcn_tensor_load_to_lds(g0, g1, z4, z4, 0);
#endif
}
#endif

// ---------------------------------------------------------------------------
// Tiled f32 WMMA GEMM:  C[M,256] = epi(A[M,KDIM] @ B[KDIM,256])
//   64x64 tile / 256-thread block, 8 waves -> 4x2 grid of 16x32 wave tiles,
//   each wave owns two N-subtiles.  K staged via TDM into double-buffered
//   LDS (A: 64x32 pad->stride 36; B: 32x64 pad->stride 68); wave 0 issues the
//   DMA for chunk i+1 while all waves compute chunk i.
// flags: bit0 = ReLU, bit1 = add residual (resid[resid_idx?resid_idx[m]:m])
// NOTE: no arrays of LDS-derived pointers (lld cannot relocate addrspacecast
// static initializers) -- buffer selection is pure arithmetic on the parity.
// ---------------------------------------------------------------------------
#define FLAG_RELU  1
#define FLAG_RESID 2

#define A_STRIDE 36
#define B_STRIDE 68
#define A_BUF_F  (64 * A_STRIDE)          // 2304 floats = 9216 B
#define B_BUF_F  (32 * B_STRIDE)          // 2176 floats = 8704 B
#define SMEM_F   (2 * (A_BUF_F + B_BUF_F))
#define SMEM_B   (SMEM_F * 4)             // 35840 bytes dynamic LDS

template <int KDIM>
__global__ __launch_bounds__(256) void gemm256(
    const float* __restrict__ A, const float* __restrict__ B,
    const float* __restrict__ bias, const float* resid,
    const int* __restrict__ resid_idx, float* C, int M, int flags)
{
    extern __shared__ float smem[];
    // byte offsets of the four LDS staging buffers (A0, A1, B0, B1)
    const unsigned ALDS0 = 0u;
    const unsigned ALDS1 = (unsigned)(A_BUF_F * 4);
    const unsigned BLDS0 = (unsigned)(2 * A_BUF_F * 4);
    const unsigned BLDS1 = (unsigned)((2 * A_BUF_F + B_BUF_F) * 4);

    const int tid  = threadIdx.x;
    const int m0   = blockIdx.x * 64;
    const int n0   = blockIdx.y * 64;
    const int lane = tid & 31;
    const int w    = tid >> 5;          // wave id, wave32
    const int wm   = w >> 1;            // 0..3
    const int wn   = w & 1;             // 0..1
    const int lh   = lane & 15;
    const bool hi  = lane >= 16;

    v8f c0 = {0.f, 0.f, 0.f, 0.f, 0.f, 0.f, 0.f, 0.f};
    v8f c1 = {0.f, 0.f, 0.f, 0.f, 0.f, 0.f, 0.f, 0.f};

    const int NCH = KDIM / 32;
    const float* Ablk = A + (size_t)m0 * KDIM;   // tile origin (m0, 0)
    const float* Bblk = B + n0;                  // tile origin (0, n0)

#if HAS_TDM
    if (tid < 32) {                     // wave 0 drives the Tensor Data Mover
        tdm_load_2d(ALDS0, Ablk, 32, 64, KDIM, 1u << 20, KDIM, 4, 3);
        tdm_load_2d(BLDS0, Bblk, 64, 32, HID,  1u << 20, HID,  5, 3);
    }
#else
    const int ar = tid >> 2, ac = (tid & 3) * 8;   // A: 64 rows x 32 cols
    const int br = tid >> 3, bc = (tid & 7) * 8;   // B: 32 rows x 64 cols
    const float* Aip = Ablk + (size_t)ar * KDIM + ac;
    const float* Bip = Bblk + (size_t)br * HID + bc;
#endif

    for (int i = 0; i < NCH; ++i) {
        const int p = i & 1;
        float* const As_ = smem + (p ? A_BUF_F : 0);
        float* const Bs_ = smem + 2 * A_BUF_F + (p ? B_BUF_F : 0);
#if HAS_TDM
        if (tid < 32) __builtin_amdgcn_s_wait_tensorcnt(0);   // chunk i landed
        __syncthreads();                 // publish chunk i; compute i-1 done
        if (tid < 32 && (i + 1) < NCH) { // DMA chunk i+1 while computing i
            tdm_load_2d(p ? ALDS0 : ALDS1, Ablk + (i + 1) * 32,
                        32, 64, KDIM, 1u << 20, KDIM, 4, 3);
            tdm_load_2d(p ? BLDS0 : BLDS1, Bblk + (size_t)(i + 1) * 32 * HID,
                        64, 32, HID, 1u << 20, HID, 5, 3);
        }
#else
        const int k0 = i * 32;
        const float4 a4a = *(const float4*)(Aip + k0);
        const float4 a4b = *(const float4*)(Aip + k0 + 4);
        const float4 b4a = *(const float4*)(Bip + (size_t)k0 * HID);
        const float4 b4b = *(const float4*)(Bip + (size_t)k0 * HID + 4);
        __syncthreads();
        *(float4*)&As_[ar * A_STRIDE + ac]     = a4a;
        *(float4*)&As_[ar * A_STRIDE + ac + 4] = a4b;
        *(float4*)&Bs_[br * B_STRIDE + bc]     = b4a;
        *(float4*)&Bs_[br * B_STRIDE + bc + 4] = b4b;
        __syncthreads();
#endif
        #pragma unroll
        for (int kk = 0; kk < 32; kk += 4) {
            const int koff = kk + (hi ? 2 : 0);
            v2f a;
            a.x = As_[(wm * 16 + lh) * A_STRIDE + koff];
            a.y = As_[(wm * 16 + lh) * A_STRIDE + koff + 1];
            v2f b0, b1;
            b0.x = Bs_[koff * B_STRIDE + wn * 32 + lh];
            b0.y = Bs_[(koff + 1) * B_STRIDE + wn * 32 + lh];
            b1.x = Bs_[koff * B_STRIDE + wn * 32 + 16 + lh];
            b1.y = Bs_[(koff + 1) * B_STRIDE + wn * 32 + 16 + lh];
            c0 = __builtin_amdgcn_wmma_f32_16x16x4_f32(
                     false, a, false, b0, (short)0, c0, false, false);
            c1 = __builtin_amdgcn_wmma_f32_16x16x4_f32(
                     false, a, false, b1, (short)0, c1, false, false);
        }
    }

    // epilogue: C/D layout -> VGPR v holds M=v (lanes 0-15) / M=v+8 (lanes 16-31)
    const int mbase = m0 + wm * 16 + (hi ? 8 : 0);
    const int nbase = n0 + wn * 32 + lh;
    #pragma unroll
    for (int v = 0; v < 8; ++v) {
        const int mr = mbase + v;
        int rr = mr;
        if ((flags & FLAG_RESID) && resid_idx) rr = resid_idx[mr];
        #pragma unroll
        for (int sub = 0; sub < 2; ++sub) {
            const int nc = nbase + sub * 16;
            float acc = (sub == 0) ? c0[v] : c1[v];
            if (bias) acc += bias[nc];
            if (flags & FLAG_RELU) acc = fmaxf(acc, 0.0f);
            if (flags & FLAG_RESID) acc += resid[(size_t)rr * HID + nc];
            C[(size_t)mr * HID + nc] = acc;
        }
    }
}

// ---------------------------------------------------------------------------
// Attention per node: its DEG=16 edges are contiguous [16i, 16i+16).
// logits -> per-head softmax over 16 -> /sqrt(64) -> aggregate V -> shifted
// softplus. One 256-thread block per node; K/V rows staged through LDS.
// ---------------------------------------------------------------------------
__global__ __launch_bounds__(256) void attn_kernel(
    const float* __restrict__ Q, const float* __restrict__ K,
    const float* __restrict__ V, float* __restrict__ act)
{
    __shared__ float Qs[HID];
    __shared__ float buf[DEG * HID];   // 16 KB, reused for K then V
    __shared__ float lg[DEG * 4];
    __shared__ float alpha[DEG * 4];

    const int i = blockIdx.x;
    const int t = threadIdx.x;

    Qs[t] = Q[(size_t)i * HID + t];
    const float* Kb = K + (size_t)i * DEG * HID;
    for (int j = t; j < DEG * HID; j += 256) buf[j] = Kb[j];
    __syncthreads();

    if (t < 64) {                       // logits: (edge e, head h)
        const int e = t >> 2, h = t & 3;
        float s = 0.0f;
        #pragma unroll
        for (int dd = 0; dd < 64; ++dd)
            s += Qs[h * 64 + dd] * buf[e * HID + h * 64 + dd];
        lg[e * 4 + h] = s;
    }
    __syncthreads();

    if (t < 4) {                        // per-head softmax over 16 edges
        const int h = t;
        float m = -3.4e38f;
        #pragma unroll
        for (int e = 0; e < DEG; ++e) m = fmaxf(m, lg[e * 4 + h]);
        float ex[DEG], ssum = 0.0f;
        #pragma unroll
        for (int e = 0; e < DEG; ++e) { ex[e] = expf(lg[e * 4 + h] - m); ssum += ex[e]; }
        const float inv = 1.0f / (ssum * 8.0f);   // softmax then / sqrt(d=64)
        #pragma unroll
        for (int e = 0; e < DEG; ++e) alpha[e * 4 + h] = ex[e] * inv;
    }
    __syncthreads();

    const float* Vb = V + (size_t)i * DEG * HID;
    for (int j = t; j < DEG * HID; j += 256) buf[j] = Vb[j];
    __syncthreads();

    const int h = t >> 6, dd = t & 63;
    float a = 0.0f;
    #pragma unroll
    for (int e = 0; e < DEG; ++e)
        a += alpha[e * 4 + h] * buf[e * HID + h * 64 + dd];
    const float sp = fmaxf(a, 0.0f) + log1pf(expf(-fabsf(a))) - LOG2F_;
    act[(size_t)i * HID + t] = sp;
}

// ---------------------------------------------------------------------------
extern "C" void kernel_launch(void* const* d_in, const int* in_sizes, int n_in,
                              void* d_out, int out_size, void* d_ws, size_t ws_size,
                              hipStream_t stream)
{
    (void)in_sizes; (void)n_in; (void)out_size; (void)ws_size;
    const float* Vraw  = (const float*)d_in[0];
    const float* Efeat = (const float*)d_in[1];
    const int*   R     = (const int*)d_in[2];
    const int*   ei    = (const int*)d_in[3];
    const float* Wv    = (const float*)d_in[4];
    const float* bv    = (const float*)d_in[5];
    const float* We    = (const float*)d_in[6];
    const float* be    = (const float*)d_in[7];
    const float* ln1w  = (const float*)d_in[8];
    const float* ln1b  = (const float*)d_in[9];
    const float* WQ    = (const float*)d_in[10];
    const float* WK    = (const float*)d_in[11];
    const float* WV    = (const float*)d_in[12];
    const float* WO    = (const float*)d_in[13];
    const float* ln2w  = (const float*)d_in[14];
    const float* ln2b  = (const float*)d_in[15];
    const float* Winw  = (const float*)d_in[16];
    const float* Winb  = (const float*)d_in[17];
    const float* Woutw = (const float*)d_in[18];
    const float* Woutb = (const float*)d_in[19];

    float* hV = (float*)d_out;                 // [NN,256] accumulator = output
    const int* col = ei + NE;                  // edge_index row 1

    // workspace layout (256B-aligned chunks)
    float* ws = (float*)d_ws;
    size_t off = 0;
    auto alloc = [&](size_t n) { float* p = ws + off; off += (n + 63) & ~(size_t)63; return p; };
    float* Vc   = alloc((size_t)NN * HID);
    float* hE   = alloc((size_t)NE * HID);
    float* yln  = alloc((size_t)NN * HID);
    float* Qb   = alloc((size_t)NN * HID);
    float* Yk   = alloc((size_t)NN * HID);
    float* Yv   = alloc((size_t)NN * HID);
    float* Kb   = alloc((size_t)NE * HID);
    float* Vb   = alloc((size_t)NE * HID);
    float* actb = alloc((size_t)NN * HID);
    float* t1   = alloc((size_t)NN * HID);

    const dim3 blk(256);
    const dim3 gN(NN / 64, HID / 64);
    const dim3 gE(NE / 64, HID / 64);

    // Vc = [V | aa_embed(R)];  hV = Vc@Wv + bv;  hE = E@We + be
    embed_kernel<<<NN, blk, 0, stream>>>(Vraw, R, Vc);
    gemm256<256><<<gN, blk, SMEM_B, stream>>>(Vc, Wv, bv, nullptr, nullptr, hV, NN, 0);
    gemm256<128><<<gE, blk, SMEM_B, stream>>>(Efeat, We, be, nullptr, nullptr, hE, NE, 0);

    for (int l = 0; l < 2; ++l) {
        const size_t sq = (size_t)l * HID * HID;        // 256x256 weights
        const size_t sk = (size_t)l * 2 * HID * HID;    // 512x256 weights
        const size_t sv = (size_t)l * HID;              // vectors

        // y = LN1(hV)
        ln_kernel<<<NN, blk, 0, stream>>>(hV, ln1w + sv, ln1b + sv, yln);
        // node-side projections
        gemm256<256><<<gN, blk, SMEM_B, stream>>>(yln, WQ + sq, nullptr, nullptr, nullptr, Qb, NN, 0);
        gemm256<256><<<gN, blk, SMEM_B, stream>>>(yln, WK + sk + (size_t)HID * HID, nullptr, nullptr, nullptr, Yk, NN, 0);
        gemm256<256><<<gN, blk, SMEM_B, stream>>>(yln, WV + sk + (size_t)HID * HID, nullptr, nullptr, nullptr, Yv, NN, 0);
        // edge-side: K = hE@WK[:256] + Yk[col],  V likewise (gathered residual)
        gemm256<256><<<gE, blk, SMEM_B, stream>>>(hE, WK + sk, nullptr, Yk, col, Kb, NE, FLAG_RESID);
        gemm256<256><<<gE, blk, SMEM_B, stream>>>(hE, WV + sk, nullptr, Yv, col, Vb, NE, FLAG_RESID);
        // segment softmax + aggregate + shifted softplus
        attn_kernel<<<NN, blk, 0, stream>>>(Qb, Kb, Vb, actb);
        // hV += act @ WO
        gemm256<256><<<gN, blk, SMEM_B, stream>>>(actb, WO + sq, nullptr, hV, nullptr, hV, NN, FLAG_RESID);
        // FFN: hV += relu(LN2(hV)@Win + b)@Wout + b
        ln_kernel<<<NN, blk, 0, stream>>>(hV, ln2w + sv, ln2b + sv, yln);
        gemm256<256><<<gN, blk, SMEM_B, stream>>>(yln, Winw + sq, Winb + sv, nullptr, nullptr, t1, NN, FLAG_RELU);
        gemm256<256><<<gN, blk, SMEM_B, stream>>>(t1, Woutw + sq, Woutb + sv, hV, nullptr, hV, NN, FLAG_RESID);
    }
}